// PNAConvPoolReadout_32195074851229
// MI455X (gfx1250) — compile-verified
//
#include <hip/hip_runtime.h>
#include <hip/hip_bf16.h>
#include <stdint.h>

// ---------------------------------------------------------------------------
// PNAConv (molecule pooling) for MI455X / gfx1250.
// f32 matrix work via V_WMMA_F32_16X16X4_F32; tile staging via
// GLOBAL_LOAD_ASYNC_TO_LDS_B128 (ASYNCcnt) with double buffering.
//
// Output = only imaginary (molecule) nodes, whose x == 0, so:
//   msg = vi @ pre_W[F:2F,:] + pre_b        (pre-MLP x_i half drops out)
//   post-MLP uses post_W rows [F:13F)       (x block multiplies zeros)
// atom_mol_batch is sorted -> contiguous segments via binary search.
// ---------------------------------------------------------------------------

typedef __attribute__((ext_vector_type(2))) float v2f;
typedef __attribute__((ext_vector_type(8))) float v8f;

// builtin's parameter type: int vector_size(16), AS1 (global) / AS3 (LDS)
typedef int v4i __attribute__((vector_size(16)));
typedef __attribute__((address_space(1))) v4i gas_v4i;   // global AS pointee
typedef __attribute__((address_space(3))) v4i las_v4i;   // LDS AS pointee

constexpr int F        = 128;
constexpr int N_ATOMS  = 65536;
constexpr int N_MOLS   = 2048;
constexpr int LDA      = 132;   // padded LDS stride for 16-row A tiles
constexpr int LDB      = 132;   // padded LDS stride for 128-row B tiles
constexpr int MOLS_PER_WG = 8;

// ---- CDNA5 async global->LDS (b128) with compile-safe fallback -------------
__device__ __forceinline__ void async_load_f4(const float* g, float* l)
{
#if defined(__has_builtin) && __has_builtin(__builtin_amdgcn_global_load_async_to_lds_b128)
    // generic LDS pointer low 32 bits == LDS offset (ISA aperture rules);
    // build AS-qualified pointers via integer casts.
    __builtin_amdgcn_global_load_async_to_lds_b128(
        (gas_v4i*)(uintptr_t)g,
        (las_v4i*)(uintptr_t)(unsigned)(uintptr_t)l,
        0, 0);
#else
    *(float4*)l = *(const float4*)g;   // synchronous fallback (same barrier protocol)
#endif
}

__device__ __forceinline__ void wait_async0()
{
#if defined(__has_builtin) && __has_builtin(__builtin_amdgcn_s_wait_asynccnt)
    __builtin_amdgcn_s_wait_asynccnt(0);
#else
    asm volatile("s_wait_asynccnt 0x0" ::: "memory");
#endif
}

// One 16x16 (M x N) f32 tile, K=128, A from LDS (row-major, stride LDA),
// B from LDS (K-major, stride LDB). pA/pB are pre-offset per-lane pointers.
// Two accumulators break the WMMA->WMMA dependency chain.
__device__ __forceinline__ v8f gemm_tile_k128(const float* __restrict__ pA,
                                              const float* __restrict__ pB,
                                              v8f acc)
{
    v8f acc2 = {};
#pragma unroll
    for (int kk = 0; kk < 128; kk += 8) {
        {
            float2 av = *(const float2*)(pA + kk);
            v2f a; a[0] = av.x; a[1] = av.y;
            v2f b; b[0] = pB[kk * LDB]; b[1] = pB[(kk + 1) * LDB];
            acc = __builtin_amdgcn_wmma_f32_16x16x4_f32(
                false, a, false, b, (short)0, acc, false, false);
        }
        {
            float2 av = *(const float2*)(pA + kk + 4);
            v2f a; a[0] = av.x; a[1] = av.y;
            v2f b; b[0] = pB[(kk + 4) * LDB]; b[1] = pB[(kk + 5) * LDB];
            acc2 = __builtin_amdgcn_wmma_f32_16x16x4_f32(
                false, a, false, b, (short)0, acc2, false, false);
        }
    }
    return acc + acc2;
}

__device__ __forceinline__ int lower_bound_i64(const long long* __restrict__ arr,
                                               int n, long long key)
{
    int lo = 0, hi = n;
    while (lo < hi) {
        int mid = (lo + hi) >> 1;
        if (arr[mid] < key) lo = mid + 1; else hi = mid;
    }
    return lo;
}

// ---------------------------------------------------------------------------
// Kernel 1: msg GEMM + fused segment reductions -> agg12 [N_MOLS, 1536]
// grid: N_MOLS/MOLS_PER_WG workgroups of 256 threads (8 wave32s = 8 N-tiles).
// dynamic LDS: sB 128*LDB + double-buffered sA 2*16*LDA floats.
// ---------------------------------------------------------------------------
__global__ __launch_bounds__(256)
void pna_msg_agg(const float* __restrict__ vi,
                 const float* __restrict__ preW2,     // pre_W + F*F : [128,128] K-major
                 const float* __restrict__ pre_b,     // [128]
                 const float* __restrict__ avg_p,     // [1]
                 const long long* __restrict__ batch, // [N_ATOMS], sorted
                 float* __restrict__ agg12)           // [N_MOLS, 1536]
{
    extern __shared__ float smem[];
    float* sB  = smem;                              // 128 x LDB (pre_W2, K-major)
    float* sA0 = smem + 128 * LDB;                  // 16 x LDA (atom chunk, buf 0)
    float* sA1 = sA0 + 16 * LDA;                    // 16 x LDA (atom chunk, buf 1)

    const int tid    = threadIdx.x;
    const int lane   = tid & 31;
    const int wave   = tid >> 5;
    const int laneLo = lane & 15;
    const bool hiHalf = lane >= 16;
    const int h2 = hiHalf ? 2 : 0;
    const int n0 = wave * 16;
    const int f  = n0 + laneLo;       // feature column owned by this lane

    // Stage pre_W2 once per workgroup (L2-resident across WGs).
    for (int q = tid; q < 128 * 32; q += 256) {
        int r = q >> 5, c4 = q & 31;
        *(float4*)(sB + r * LDB + c4 * 4) =
            *(const float4*)(preW2 + r * 128 + c4 * 4);
    }

    const float bval = pre_b[f];
    const float avgv = avg_p[0];

    for (int mi = 0; mi < MOLS_PER_WG; ++mi) {
        const int m     = blockIdx.x * MOLS_PER_WG + mi;
        const int start = lower_bound_i64(batch, N_ATOMS, (long long)m);
        const int e     = lower_bound_i64(batch, N_ATOMS, (long long)(m + 1));
        const int cnt   = e - start;
        const int nch   = (cnt + 15) >> 4;

        float S = 0.f, Q = 0.f;
        float MN = __builtin_inff(), MX = -__builtin_inff();

        if (nch > 0) {
            __syncthreads();   // all waves done with both A buffers (and sB staged)
            // issue chunk 0 -> buf 0 (tail rows left stale: masked in reduction)
            for (int q = tid; q < 512; q += 256) {
                int r = q >> 5, c4 = q & 31;
                if (r < cnt)
                    async_load_f4(vi + (size_t)(start + r) * F + c4 * 4,
                                  sA0 + r * LDA + c4 * 4);
            }
        }

        for (int c = 0; c < nch; ++c) {
            const int base = c * 16;
            wait_async0();     // this wave's issued async loads have landed
            __syncthreads();   // all waves' chunk data visible; prior compute done

            if (c + 1 < nch) { // issue chunk c+1 into the other buffer
                const int nb = base + 16;
                float* dst = ((c + 1) & 1) ? sA1 : sA0;
                for (int q = tid; q < 512; q += 256) {
                    int r = q >> 5, c4 = q & 31;
                    if (nb + r < cnt)
                        async_load_f4(vi + (size_t)(start + nb + r) * F + c4 * 4,
                                      dst + r * LDA + c4 * 4);
                }
            }

            const float* cA = (c & 1) ? sA1 : sA0;
            v8f acc = {};
            acc = gemm_tile_k128(cA + laneLo * LDA + h2,
                                 sB + h2 * LDB + f, acc);

            // fused reduction of this 16-row tile (rows = atoms, cols = features)
            float cs = 0.f, cq = 0.f;
            float cmn = __builtin_inff(), cmx = -__builtin_inff();
#pragma unroll
            for (int r = 0; r < 8; ++r) {
                int mrow = r + (hiHalf ? 8 : 0);
                float v = acc[r] + bval;               // add pre_b
                if (base + mrow < cnt) {               // masks stale tail rows too
                    cs += v; cq += v * v;
                    cmn = fminf(cmn, v); cmx = fmaxf(cmx, v);
                }
            }
            // fold the two half-wave row groups (same feature column)
            cs  += __shfl_xor(cs, 16, 32);
            cq  += __shfl_xor(cq, 16, 32);
            cmn  = fminf(cmn, __shfl_xor(cmn, 16, 32));
            cmx  = fmaxf(cmx, __shfl_xor(cmx, 16, 32));
            S += cs; Q += cq;
            MN = fminf(MN, cmn); MX = fmaxf(MX, cmx);
        }

        if (lane < 16) {
            const float cntf  = (float)cnt;
            const float safe  = fmaxf(cntf, 1.f);
            const float mean  = S / safe;
            const float mean2 = Q / safe;
            const float var   = fmaxf(mean2 - mean * mean, 0.f);
            const float stdv  = sqrtf(var + 1e-5f);
            const float mnv   = (cnt > 0) ? MN : 0.f;
            const float mxv   = (cnt > 0) ? MX : 0.f;
            const float logd  = logf(safe + 1.f);
            const float sa    = logd / avgv;   // amplification scaler
            const float sb    = avgv / logd;   // attenuation scaler

            float* row = agg12 + (size_t)m * 1536;
            row[        f] = mean;      row[ 128 + f] = mnv;
            row[ 256 + f] = mxv;        row[ 384 + f] = stdv;
            row[ 512 + f] = mean * sa;  row[ 640 + f] = mnv * sa;
            row[ 768 + f] = mxv * sa;   row[ 896 + f] = stdv * sa;
            row[1024 + f] = mean * sb;  row[1152 + f] = mnv * sb;
            row[1280 + f] = mxv * sb;   row[1408 + f] = stdv * sb;
        }
    }
}

// ---------------------------------------------------------------------------
// Kernel 2: out = (agg12 @ postW2 + post_b) @ lin_W + lin_b -> d_out [2048,128]
// grid: N_MOLS/16 workgroups; 16-molecule M-tile, 8 waves = 8 N-tiles.
// K=1536 streamed as 12 double-buffered async-staged chunks of 128.
// dynamic LDS: 2*(128*LDB) + 2*(16*LDA) + 16*LDA floats (~157 KB).
// ---------------------------------------------------------------------------
__global__ __launch_bounds__(256)
void pna_post(const float* __restrict__ agg12,   // [N_MOLS, 1536]
              const float* __restrict__ postW2,  // post_W + F*F : [1536,128] K-major
              const float* __restrict__ post_b,  // [128]
              const float* __restrict__ linW,    // [128,128] K-major
              const float* __restrict__ lin_b,   // [128]
              float* __restrict__ out)           // [N_MOLS, 128]
{
    extern __shared__ float smem[];
    float* sB0 = smem;                                     // 128 x LDB
    float* sB1 = smem + 128 * LDB;                         // 128 x LDB
    float* sA0 = smem + 2 * 128 * LDB;                     // 16 x LDA
    float* sA1 = smem + 2 * 128 * LDB + 16 * LDA;          // 16 x LDA
    float* sC  = smem + 2 * 128 * LDB + 2 * 16 * LDA;      // 16 x LDA

    const int tid    = threadIdx.x;
    const int lane   = tid & 31;
    const int wave   = tid >> 5;
    const int laneLo = lane & 15;
    const bool hiHalf = lane >= 16;
    const int h2 = hiHalf ? 2 : 0;
    const int n0 = wave * 16;
    const int f  = n0 + laneLo;
    const int mt = blockIdx.x * 16;

    // issue one K-chunk (A tile + B tile) into buffer bi
    auto issue_chunk = [&](int kc, int bi) {
        float* dA = bi ? sA1 : sA0;
        float* dB = bi ? sB1 : sB0;
        for (int q = tid; q < 512; q += 256) {         // A: 16 x 128
            int r = q >> 5, c4 = q & 31;
            async_load_f4(agg12 + (size_t)(mt + r) * 1536 + kc * 128 + c4 * 4,
                          dA + r * LDA + c4 * 4);
        }
        for (int q = tid; q < 4096; q += 256) {        // B: 128 x 128
            int r = q >> 5, c4 = q & 31;
            async_load_f4(postW2 + (size_t)(kc * 128 + r) * 128 + c4 * 4,
                          dB + r * LDB + c4 * 4);
        }
    };

    issue_chunk(0, 0);

    v8f acc = {};
    for (int kc = 0; kc < 12; ++kc) {
        wait_async0();     // this wave's async loads for chunk kc are in LDS
        __syncthreads();   // everyone's data visible; prior compute on buf done
        if (kc + 1 < 12) issue_chunk(kc + 1, (kc + 1) & 1);
        const float* cA = (kc & 1) ? sA1 : sA0;
        const float* cB = (kc & 1) ? sB1 : sB0;
        acc = gemm_tile_k128(cA + laneLo * LDA + h2,
                             cB + h2 * LDB + f, acc);
    }

    // out1 = acc + post_b -> LDS (becomes A of the second GEMM)
    const float pb = post_b[f];
#pragma unroll
    for (int r = 0; r < 8; ++r)
        sC[(r + (hiHalf ? 8 : 0)) * LDA + f] = acc[r] + pb;

    // stage lin_W into sB buffer 0 (last read of buf0 finished at kc==10,
    // guaranteed by the barrier at top of kc==11)
    for (int q = tid; q < 4096; q += 256) {
        int r = q >> 5, c4 = q & 31;
        *(float4*)(sB0 + r * LDB + c4 * 4) =
            *(const float4*)(linW + r * 128 + c4 * 4);
    }
    __syncthreads();

    v8f acc2 = {};
    acc2 = gemm_tile_k128(sC + laneLo * LDA + h2, sB0 + h2 * LDB + f, acc2);

    const float lb = lin_b[f];
#pragma unroll
    for (int r = 0; r < 8; ++r)
        out[(size_t)(mt + r + (hiHalf ? 8 : 0)) * 128 + f] = acc2[r] + lb;
}

// ---------------------------------------------------------------------------
extern "C" void kernel_launch(void* const* d_in, const int* in_sizes, int n_in,
                              void* d_out, int out_size, void* d_ws, size_t ws_size,
                              hipStream_t stream)
{
    (void)in_sizes; (void)n_in; (void)out_size; (void)ws_size;

    const float*     vi      = (const float*)d_in[0];
    const float*     pre_W   = (const float*)d_in[1];
    const float*     pre_b   = (const float*)d_in[2];
    const float*     post_W  = (const float*)d_in[3];
    const float*     post_b  = (const float*)d_in[4];
    const float*     lin_W   = (const float*)d_in[5];
    const float*     lin_b   = (const float*)d_in[6];
    const float*     avg_log = (const float*)d_in[7];
    const long long* batch   = (const long long*)d_in[8];
    float*           out     = (float*)d_out;
    float*           agg12   = (float*)d_ws;          // [N_MOLS, 1536] f32 = 12.6 MB

    const size_t sm1 = (size_t)(128 * LDB + 2 * 16 * LDA) * sizeof(float);
    const size_t sm2 = (size_t)(2 * 128 * LDB + 3 * 16 * LDA) * sizeof(float);

    pna_msg_agg<<<N_MOLS / MOLS_PER_WG, 256, sm1, stream>>>(
        vi, pre_W + F * F, pre_b, avg_log, batch, agg12);

    pna_post<<<N_MOLS / 16, 256, sm2, stream>>>(
        agg12, post_W + F * F, post_b, lin_W, lin_b, out);
}